// TNet_53102975648413
// MI455X (gfx1250) — compile-verified
//
#include <hip/hip_runtime.h>
#include <hip/hip_bf16.h>

// ---------------------------------------------------------------------------
// TNet (DGCNN transform net), B=8 C=3 N=4096 K=20.
//  k_init      : zero BN accumulators, +-inf global max/min buffers
//  k_knn       : per-point top-20 nearest neighbors (LDS-tiled)
//  k_stats1    : sum/sumsq of z1 = edge@w1^T over (B,N,K) per 64 ch
//  k_finalize  : BN stats -> (scale, shift) per channel
//  k_conv2     : h1 (20x64, zero-padded to 32) -> z2 = h1@w2^T via
//                V_WMMA_F32_16X16X4_F32 (2 M-tiles x 16 K-steps per wave),
//                per-(b,n,ch) max/min over K + BN2 sum/sumsq (padding masked)
//  k_conv3     : WMMA f32 GEMM (32768x1024x128): hmax@w3^T, atomic
//                per-channel sum/sumsq + per-(b,ch) max/min
//  k_tail      : BN3 finalize + monotone max trick; FC4/FC5/final + eye(3)
// ---------------------------------------------------------------------------

typedef __attribute__((ext_vector_type(2))) float v2f;
typedef __attribute__((ext_vector_type(8))) float v8f;

#define NPTS   4096
#define NB     8
#define KNN    20
#define EPS    1e-5f

__device__ __forceinline__ void atomicMaxF(float* addr, float v) {
    if (v >= 0.0f) atomicMax((int*)addr, __float_as_int(v));
    else           atomicMin((unsigned int*)addr, __float_as_uint(v));
}
__device__ __forceinline__ void atomicMinF(float* addr, float v) {
    if (v >= 0.0f) atomicMin((int*)addr, __float_as_int(v));
    else           atomicMax((unsigned int*)addr, __float_as_uint(v));
}

// ---------------------------------------------------------------------------
__global__ __launch_bounds__(256)
void k_init(float* acc, float* zmax3, float* zmin3) {
    int t = blockIdx.x * 256 + threadIdx.x;
    if (t < 2432) acc[t] = 0.0f;
    if (t < 8192) {
        zmax3[t] = -__builtin_inff();
        zmin3[t] =  __builtin_inff();
    }
}

// ---------------------------------------------------------------------------
// kNN: one thread per point, 256-point LDS tiles of candidates.
__global__ __launch_bounds__(256)
void k_knn(const float* __restrict__ x, int* __restrict__ idx) {
    __shared__ float4 tile[256];
    const int b = blockIdx.y;
    const int n = blockIdx.x * 256 + threadIdx.x;
    const float* xb = x + (long)b * 3 * NPTS;
    const float px = xb[n], py = xb[NPTS + n], pz = xb[2 * NPTS + n];
    const float psq = px * px + py * py + pz * pz;

    float bd[KNN]; int bi[KNN];
#pragma unroll
    for (int j = 0; j < KNN; ++j) { bd[j] = __builtin_inff(); bi[j] = 0; }

    for (int t0 = 0; t0 < NPTS; t0 += 256) {
        int m = t0 + threadIdx.x;
        float qx = xb[m], qy = xb[NPTS + m], qz = xb[2 * NPTS + m];
        tile[threadIdx.x] = make_float4(qx, qy, qz, qx * qx + qy * qy + qz * qz);
        __syncthreads();
        for (int i = 0; i < 256; ++i) {
            float4 q = tile[i];
            float d = psq + q.w - 2.0f * (px * q.x + py * q.y + pz * q.z);
            if (d < bd[KNN - 1]) {           // stable sorted insert
                float cd = d; int ci = t0 + i;
#pragma unroll
                for (int j = 0; j < KNN; ++j) {
                    if (cd < bd[j]) {
                        float tf = bd[j]; bd[j] = cd; cd = tf;
                        int   ti = bi[j]; bi[j] = ci; ci = ti;
                    }
                }
            }
        }
        __syncthreads();
    }
    long base = ((long)b * NPTS + n) * KNN;
#pragma unroll
    for (int j = 0; j < KNN; ++j) idx[base + j] = bi[j];
}

// ---------------------------------------------------------------------------
// BN1 statistics over z1 = edge @ w1^T.
__global__ __launch_bounds__(256)
void k_stats1(const float* __restrict__ x, const int* __restrict__ idx,
              const float* __restrict__ w1, float* sum1, float* sumsq1) {
    __shared__ float w1s[64 * 6];
    __shared__ float ps[4][64];
    __shared__ float pq[4][64];
    const int tid = threadIdx.x;
    const int o = tid & 63, slot = tid >> 6;
    for (int e = tid; e < 384; e += 256) w1s[e] = w1[e];
    __syncthreads();

    float s = 0.0f, q = 0.0f;
    long r0 = (long)blockIdx.x * 128 + (long)slot * 32;
    const float* wr = &w1s[o * 6];
    for (int i = 0; i < 32; ++i) {
        long r = r0 + i;
        long bn = r / KNN;
        int n = (int)(bn & (NPTS - 1));
        int b = (int)(bn >> 12);
        int m = idx[r];
        const float* xb = x + (long)b * 3 * NPTS;
        float cx = xb[n], cy = xb[NPTS + n], cz = xb[2 * NPTS + n];
        float nx = xb[m], ny = xb[NPTS + m], nz = xb[2 * NPTS + m];
        float z = cx * wr[0] + cy * wr[1] + cz * wr[2]
                + (nx - cx) * wr[3] + (ny - cy) * wr[4] + (nz - cz) * wr[5];
        s += z; q += z * z;
    }
    ps[slot][o] = s; pq[slot][o] = q;
    __syncthreads();
    if (tid < 64) {
        float S = ps[0][tid] + ps[1][tid] + ps[2][tid] + ps[3][tid];
        float Q = pq[0][tid] + pq[1][tid] + pq[2][tid] + pq[3][tid];
        atomicAdd(&sum1[tid], S);
        atomicAdd(&sumsq1[tid], Q);
    }
}

// ---------------------------------------------------------------------------
__global__ void k_finalize(const float* sum, const float* sumsq,
                           const float* g, const float* bb,
                           float* scale, float* shift, float invCount) {
    int t = blockIdx.x * blockDim.x + threadIdx.x;
    float m = sum[t] * invCount;
    float v = sumsq[t] * invCount - m * m;
    float sc = g[t] * rsqrtf(v + EPS);
    scale[t] = sc;
    shift[t] = bb[t] - m * sc;
}

// ---------------------------------------------------------------------------
// Fused conv1(+BN1+relu) -> conv2 via WMMA. Block = one (b,n), 256 threads
// (8 waves). h1 zero-padded 20->32 rows; wave w owns columns [w*16,w*16+16)
// with two M-tiles (rows 0-15, 16-31). Padding rows are masked out of stats.
__global__ __launch_bounds__(256)
void k_conv2(const float* __restrict__ x, const int* __restrict__ idx,
             const float* __restrict__ w1, const float* __restrict__ scale1,
             const float* __restrict__ shift1, const float* __restrict__ w2,
             float* __restrict__ zmax2, float* __restrict__ zmin2,
             float* sum2, float* sumsq2) {
    __shared__ float h1p[32][64];     // 8 KB, rows 20..31 zero
    __shared__ float nb[KNN][3];
    __shared__ float ctr[3];
    __shared__ float w1s[64 * 6];
    __shared__ float sc1[64], sh1[64];
    const int tid = threadIdx.x;
    const long bn = blockIdx.x;
    const int n = (int)(bn & (NPTS - 1));
    const int b = (int)(bn >> 12);
    const float* xb = x + (long)b * 3 * NPTS;

    for (int e = tid; e < 384; e += 256) w1s[e] = w1[e];
    if (tid < 64) { sc1[tid] = scale1[tid]; sh1[tid] = shift1[tid]; }
    if (tid < KNN) {
        int m = idx[bn * KNN + tid];
        nb[tid][0] = xb[m]; nb[tid][1] = xb[NPTS + m]; nb[tid][2] = xb[2 * NPTS + m];
    }
    if (tid < 3) ctr[tid] = xb[tid * NPTS + n];
    __syncthreads();

    for (int e = tid; e < 32 * 64; e += 256) {
        int k = e >> 6, o = e & 63;
        float hv = 0.0f;
        if (k < KNN) {
            const float* wr = &w1s[o * 6];
            float z = ctr[0] * wr[0] + ctr[1] * wr[1] + ctr[2] * wr[2]
                    + (nb[k][0] - ctr[0]) * wr[3]
                    + (nb[k][1] - ctr[1]) * wr[4]
                    + (nb[k][2] - ctr[2]) * wr[5];
            float h = z * sc1[o] + sh1[o];
            hv = h > 0.0f ? h : 0.0f;
        }
        h1p[k][o] = hv;
    }
    __syncthreads();

    const int wv = tid >> 5;          // 16-column subtile
    const int lane = tid & 31;
    const int lm = lane & 15;
    const int hi = (lane >> 4) << 1;  // 0 or 2

    // Preload the 16 B fragments (w2 row segment) into registers; w2 is
    // (128,64) row-major, i.e. already [col][k].
    const float* brow = w2 + (long)(wv * 16 + lm) * 64;
    v2f bf[16];
#pragma unroll
    for (int ks = 0; ks < 16; ++ks) {
        bf[ks].x = brow[ks * 4 + hi];
        bf[ks].y = brow[ks * 4 + hi + 1];
    }

    const float* a0row = &h1p[lm][0];
    const float* a1row = &h1p[16 + lm][0];
    v8f acc0 = {}, acc1 = {};
#pragma unroll
    for (int ks = 0; ks < 16; ++ks) {
        v2f a0; a0.x = a0row[ks * 4 + hi]; a0.y = a0row[ks * 4 + hi + 1];
        v2f a1; a1.x = a1row[ks * 4 + hi]; a1.y = a1row[ks * 4 + hi + 1];
        acc0 = __builtin_amdgcn_wmma_f32_16x16x4_f32(false, a0, false, bf[ks],
                                                     (short)0, acc0, false, false);
        acc1 = __builtin_amdgcn_wmma_f32_16x16x4_f32(false, a1, false, bf[ks],
                                                     (short)0, acc1, false, false);
    }

    // Stats over the 20 valid rows of this lane's column.
    // acc0: rows 0-7 (lanes<16) / 8-15 (lanes>=16): all valid.
    // acc1: rows 16-23 (lanes<16) -> acc1[0..3] = rows 16-19 valid;
    //       rows 24-31 (lanes>=16): all padding.
    float s = 0.0f, q = 0.0f, mx = -__builtin_inff(), mn = __builtin_inff();
#pragma unroll
    for (int i = 0; i < 8; ++i) {
        float v = acc0[i];
        s += v; q += v * v;
        mx = fmaxf(mx, v); mn = fminf(mn, v);
    }
    s  += __shfl_xor(s, 16, 32);
    q  += __shfl_xor(q, 16, 32);
    mx = fmaxf(mx, __shfl_xor(mx, 16, 32));
    mn = fminf(mn, __shfl_xor(mn, 16, 32));
    if (lane < 16) {
#pragma unroll
        for (int i = 0; i < 4; ++i) {
            float v = acc1[i];
            s += v; q += v * v;
            mx = fmaxf(mx, v); mn = fminf(mn, v);
        }
        int col = wv * 16 + lm;
        zmax2[bn * 128 + col] = mx;
        zmin2[bn * 128 + col] = mn;
        atomicAdd(&sum2[col], s);
        atomicAdd(&sumsq2[col], q);
    }
}

// ---------------------------------------------------------------------------
// conv3: hmax(32768 x 128) @ w3^T(128 x 1024) via V_WMMA_F32_16X16X4_F32.
__global__ __launch_bounds__(256)
void k_conv3(const float* __restrict__ zmax2, const float* __restrict__ zmin2,
             const float* __restrict__ scale2, const float* __restrict__ shift2,
             const float* __restrict__ w3,
             float* sum3, float* sumsq3, float* zmax3, float* zmin3) {
    __shared__ float A[16][128];      // 8 KB  : 16 rows of hmax
    __shared__ float Bt[128][128];    // 64 KB : w3 cols [cg*128, cg*128+128)
    const int rt = blockIdx.x;        // 0..2047
    const int cg = blockIdx.y;        // 0..7
    const int tid = threadIdx.x;

    __builtin_prefetch(w3 + (long)cg * 128 * 128, 0, 0);   // global_prefetch_b8

    for (int e = tid; e < 16 * 128; e += 256) {
        int r = e >> 7, c = e & 127;
        long row = (long)rt * 16 + r;
        float sc = scale2[c], sh = shift2[c];
        float z = (sc >= 0.0f) ? zmax2[row * 128 + c] : zmin2[row * 128 + c];
        float h = sc * z + sh;
        A[r][c] = h > 0.0f ? h : 0.0f;
    }
    for (int e = tid; e < 128 * 32; e += 256) {            // float4 loads of w3 rows
        int col = e >> 5, qd = e & 31;
        const float4 v = ((const float4*)(w3 + ((long)cg * 128 + col) * 128))[qd];
        Bt[col][qd * 4 + 0] = v.x; Bt[col][qd * 4 + 1] = v.y;
        Bt[col][qd * 4 + 2] = v.z; Bt[col][qd * 4 + 3] = v.w;
    }
    __syncthreads();

    const int wv = tid >> 5;
    const int lane = tid & 31;
    const int lm = lane & 15;
    const int hi = (lane >> 4) << 1;
    const float* Arow = &A[lm][0];
    const float* Brow = &Bt[wv * 16 + lm][0];

    v8f acc = {};
#pragma unroll
    for (int k = 0; k < 128; k += 4) {
        v2f a; a.x = Arow[k + hi]; a.y = Arow[k + hi + 1];
        v2f bfr; bfr.x = Brow[k + hi]; bfr.y = Brow[k + hi + 1];
        acc = __builtin_amdgcn_wmma_f32_16x16x4_f32(false, a, false, bfr,
                                                    (short)0, acc, false, false);
    }

    float s = 0.0f, q2 = 0.0f, mx = -__builtin_inff(), mn = __builtin_inff();
#pragma unroll
    for (int i = 0; i < 8; ++i) {
        float v = acc[i];
        s += v; q2 += v * v;
        mx = fmaxf(mx, v); mn = fminf(mn, v);
    }
    s  += __shfl_xor(s, 16, 32);
    q2 += __shfl_xor(q2, 16, 32);
    mx = fmaxf(mx, __shfl_xor(mx, 16, 32));
    mn = fminf(mn, __shfl_xor(mn, 16, 32));
    if (lane < 16) {
        int col = cg * 128 + wv * 16 + lm;
        int b = (rt * 16) >> 12;
        atomicAdd(&sum3[col], s);
        atomicAdd(&sumsq3[col], q2);
        atomicMaxF(&zmax3[b * 1024 + col], mx);
        atomicMinF(&zmin3[b * 1024 + col], mn);
    }
}

// ---------------------------------------------------------------------------
// Tail: BN3 finalize + global max -> gmax(8,1024); FC4/FC5; final + eye(3).
__global__ __launch_bounds__(1024)
void k_tail(const float* sum3, const float* sumsq3,
            const float* zmax3, const float* zmin3,
            const float* __restrict__ g3, const float* __restrict__ b3,
            const float* __restrict__ w4, const float* __restrict__ g4,
            const float* __restrict__ b4,
            const float* __restrict__ w5, const float* __restrict__ g5,
            const float* __restrict__ b5,
            const float* __restrict__ wl, const float* __restrict__ bl,
            float* __restrict__ out) {
    __shared__ float G[NB][1024];
    __shared__ float H4[NB][512];
    __shared__ float H5[NB][256];
    const int t = threadIdx.x;
    {
        const float inv = 1.0f / (float)(NB * NPTS);
        float m = sum3[t] * inv;
        float v = sumsq3[t] * inv - m * m;
        float sc = g3[t] * rsqrtf(v + EPS);
        float sh = b3[t] - m * sc;
        for (int b = 0; b < NB; ++b) {
            float z = (sc >= 0.0f) ? zmax3[b * 1024 + t] : zmin3[b * 1024 + t];
            float h = z * sc + sh;
            G[b][t] = h > 0.0f ? h : 0.0f;
        }
    }
    __syncthreads();
    if (t < 512) {
        float z[NB];
        const float* wr = w4 + (long)t * 1024;
        for (int b = 0; b < NB; ++b) {
            float a = 0.0f;
            for (int c = 0; c < 1024; ++c) a += G[b][c] * wr[c];
            z[b] = a;
        }
        float m = 0.0f;
        for (int b = 0; b < NB; ++b) m += z[b];
        m *= (1.0f / NB);
        float v = 0.0f;
        for (int b = 0; b < NB; ++b) { float d = z[b] - m; v += d * d; }
        v *= (1.0f / NB);
        float sc = g4[t] * rsqrtf(v + EPS);
        float sh = b4[t] - m * sc;
        for (int b = 0; b < NB; ++b) {
            float h = z[b] * sc + sh;
            H4[b][t] = h > 0.0f ? h : 0.0f;
        }
    }
    __syncthreads();
    if (t < 256) {
        float z[NB];
        const float* wr = w5 + (long)t * 512;
        for (int b = 0; b < NB; ++b) {
            float a = 0.0f;
            for (int c = 0; c < 512; ++c) a += H4[b][c] * wr[c];
            z[b] = a;
        }
        float m = 0.0f;
        for (int b = 0; b < NB; ++b) m += z[b];
        m *= (1.0f / NB);
        float v = 0.0f;
        for (int b = 0; b < NB; ++b) { float d = z[b] - m; v += d * d; }
        v *= (1.0f / NB);
        float sc = g5[t] * rsqrtf(v + EPS);
        float sh = b5[t] - m * sc;
        for (int b = 0; b < NB; ++b) {
            float h = z[b] * sc + sh;
            H5[b][t] = h > 0.0f ? h : 0.0f;
        }
    }
    __syncthreads();
    if (t < NB * 9) {
        int b = t / 9, j = t % 9;
        float a = bl[j];
        const float* wr = wl + (long)j * 256;
        for (int c = 0; c < 256; ++c) a += H5[b][c] * wr[c];
        if (j == 0 || j == 4 || j == 8) a += 1.0f;
        out[b * 9 + j] = a;
    }
}

// ---------------------------------------------------------------------------
extern "C" void kernel_launch(void* const* d_in, const int* in_sizes, int n_in,
                              void* d_out, int out_size, void* d_ws, size_t ws_size,
                              hipStream_t stream) {
    const float* x  = (const float*)d_in[0];
    const float* w1 = (const float*)d_in[1];
    const float* g1 = (const float*)d_in[2];
    const float* b1 = (const float*)d_in[3];
    const float* w2 = (const float*)d_in[4];
    const float* g2 = (const float*)d_in[5];
    const float* b2 = (const float*)d_in[6];
    const float* w3 = (const float*)d_in[7];
    const float* g3 = (const float*)d_in[8];
    const float* b3 = (const float*)d_in[9];
    const float* w4 = (const float*)d_in[10];
    const float* g4 = (const float*)d_in[11];
    const float* b4 = (const float*)d_in[12];
    const float* w5 = (const float*)d_in[13];
    const float* g5 = (const float*)d_in[14];
    const float* b5 = (const float*)d_in[15];
    const float* wl = (const float*)d_in[16];
    const float* bl = (const float*)d_in[17];
    float* out = (float*)d_out;

    // workspace layout (floats)
    float* ws = (float*)d_ws;
    float* sum1   = ws + 0;          // 64
    float* sumsq1 = ws + 64;         // 64
    float* sum2   = ws + 128;        // 128
    float* sumsq2 = ws + 256;        // 128
    float* sum3   = ws + 384;        // 1024
    float* sumsq3 = ws + 1408;       // 1024   (zero-init zone = 2432 floats)
    float* scale1 = ws + 2432;       // 64
    float* shift1 = ws + 2496;       // 64
    float* scale2 = ws + 2560;       // 128
    float* shift2 = ws + 2688;       // 128
    float* zmax3  = ws + 2816;       // 8192
    float* zmin3  = ws + 11008;      // 8192
    int*   idx    = (int*)(ws + 19200);           // 8*4096*20 = 655360
    float* zmax2  = ws + 19200 + 655360;          // 32768*128
    float* zmin2  = zmax2 + 32768 * 128;          // 32768*128
    // total = 9,063,168 floats = ~36.3 MB

    k_init<<<32, 256, 0, stream>>>(ws, zmax3, zmin3);
    k_knn<<<dim3(NPTS / 256, NB), 256, 0, stream>>>(x, idx);
    k_stats1<<<(NB * NPTS * KNN) / 128, 256, 0, stream>>>(x, idx, w1, sum1, sumsq1);
    k_finalize<<<1, 64, 0, stream>>>(sum1, sumsq1, g1, b1, scale1, shift1,
                                     1.0f / (float)(NB * NPTS * KNN));
    k_conv2<<<NB * NPTS, 256, 0, stream>>>(x, idx, w1, scale1, shift1, w2,
                                           zmax2, zmin2, sum2, sumsq2);
    k_finalize<<<1, 128, 0, stream>>>(sum2, sumsq2, g2, b2, scale2, shift2,
                                      1.0f / (float)(NB * NPTS * KNN));
    k_conv3<<<dim3((NB * NPTS) / 16, 8), 256, 0, stream>>>(zmax2, zmin2, scale2, shift2,
                                                           w3, sum3, sumsq3, zmax3, zmin3);
    k_tail<<<1, 1024, 0, stream>>>(sum3, sumsq3, zmax3, zmin3, g3, b3,
                                   w4, g4, b4, w5, g5, b5, wl, bl, out);
    (void)in_sizes; (void)n_in; (void)out_size; (void)ws_size;
}